// AdaptiveGraphConv_9457517986422
// MI455X (gfx1250) — compile-verified
//
#include <hip/hip_runtime.h>

// ---------------- problem constants ----------------
#define NNODES 325
#define NP     336            // nodes padded to 21*16
#define LPS    338            // LDS row stride for L panel (bank-conflict-free: 18m mod 64)
#define ZTS    68             // LDS row stride for Z tile  (bank-conflict-free: 4m mod 64)
#define CCH    64
#define TT     12
#define BB     64
#define BT     (BB*TT)        // 768 slabs
#define MT     (NP/16)        // 21 row tiles

typedef float v2f __attribute__((ext_vector_type(2)));
typedef float v8f __attribute__((ext_vector_type(8)));

// D = A(16x4,f32) * B(4x16,f32) + C(16x16,f32)  -- CDNA5 fp32 WMMA
static __device__ __forceinline__ v8f wmma4(v2f a, v2f b, v8f c) {
    return __builtin_amdgcn_wmma_f32_16x16x4_f32(
        /*neg_a=*/false, a, /*neg_b=*/false, b,
        /*c_mod=*/(short)0, c, /*reuse_a=*/false, /*reuse_b=*/false);
}

static __device__ __forceinline__ v2f lds_pair(const float* p) {
    return *(const v2f*)p;    // 8B-aligned by construction -> ds_load_b64
}

// ---------------- prep: degree -> D^{-1/2} ----------------
__global__ void k_deg(const float* __restrict__ adj, float* __restrict__ dinv) {
    int i = blockIdx.x * blockDim.x + threadIdx.x;
    if (i < NNODES) {
        float s = 0.f;
        for (int j = 0; j < NNODES; ++j) s += adj[i * NNODES + j];
        dinv[i] = (s > 0.f) ? rsqrtf(s) : 0.f;
    }
}

// ---------------- prep: padded normalized Laplacian ----------------
__global__ void k_lap(const float* __restrict__ adj, const float* __restrict__ dinv,
                      float* __restrict__ Lp) {
    int i = blockIdx.x;        // 0..NP-1 (row)
    int j = threadIdx.x;       // 0..351
    if (j >= NP) return;
    float v = 0.f;
    if (i < NNODES && j < NNODES) {
        v = ((i == j) ? 1.f : 0.f) - dinv[i] * adj[i * NNODES + j] * dinv[j];
    }
    Lp[i * NP + j] = v;
}

// ---------------- prep: weight pack {W0-W2, W1, 2*W2} ----------------
__global__ void k_wprep(const float* __restrict__ W, float* __restrict__ Wm) {
    int i = blockIdx.x * blockDim.x + threadIdx.x;
    if (i < CCH * CCH) {
        float w0 = W[i], w1 = W[CCH*CCH + i], w2 = W[2*CCH*CCH + i];
        Wm[i]             = w0 - w2;   // folds the "-X*W2" term of Tx2
        Wm[CCH*CCH + i]   = w1;
        Wm[2*CCH*CCH + i] = 2.f * w2;  // folds the "2*" of Tx2
    }
}

// ---------------- transpose x[B,C,N,T] -> Xr[(b,t)][n pad 336][c] ----------------
__global__ void k_xpose(const float* __restrict__ x, float* __restrict__ Xr) {
    __shared__ float tile[CCH * TT];   // 768 floats
    int b = blockIdx.x;                // 0..63
    int n = blockIdx.y;                // 0..335
    int tid = threadIdx.x;             // 256 threads
    if (n < NNODES) {
        for (int i = tid; i < CCH * TT; i += blockDim.x) {
            int c = i / TT, t = i % TT;                        // reads: 48B runs over t
            tile[i] = x[((size_t)(b * CCH + c) * NNODES + n) * TT + t];
        }
        __syncthreads();
        for (int i = tid; i < CCH * TT; i += blockDim.x) {
            int c = i & 63, t = i >> 6;                        // writes: 256B runs over c
            Xr[((size_t)(b * TT + t) * NP + n) * CCH + c] = tile[c * TT + t];
        }
    } else {
        for (int i = tid; i < CCH * TT; i += blockDim.x) {
            int c = i & 63, t = i >> 6;
            Xr[((size_t)(b * TT + t) * NP + n) * CCH + c] = 0.f;
        }
    }
}

// ---------------- GEMM1: Y1 = L * Xr  (per slab) ----------------
__global__ __launch_bounds__(128)
void k_gemm_LX(const float* __restrict__ Lp, const float* __restrict__ Xr,
               float* __restrict__ Y1) {
    __shared__ float Lpan[16 * LPS];             // padded stride: conflict-free A feed
    int mtile = blockIdx.x;                      // 0..20
    int slab  = blockIdx.y;                      // 0..767
    int tid  = threadIdx.x;
    int wave = tid >> 5;                         // 4 waves: one c-tile each
    int lane = tid & 31;

    for (int i = tid; i < 16 * NP; i += 128)
        Lpan[(i / NP) * LPS + (i % NP)] = Lp[(mtile * 16 + i / NP) * NP + (i % NP)];
    __syncthreads();

    const float* Xs = Xr + (size_t)slab * NP * CCH;
    int m   = lane & 15;                         // A row
    int kh  = (lane >> 4) * 2;                   // per-lane K offset
    int col = wave * 16 + (lane & 15);           // B/D column

    v8f acc = {0.f,0.f,0.f,0.f,0.f,0.f,0.f,0.f};
    #pragma unroll 4
    for (int k = 0; k < NP; k += 4) {
        v2f a  = lds_pair(&Lpan[m * LPS + k + kh]);
        v2f bb;
        bb.x = Xs[(k + kh) * CCH + col];         // coalesced 64B/instr across lanes
        bb.y = Xs[(k + kh + 1) * CCH + col];
        acc  = wmma4(a, bb, acc);
    }

    float* Ys = Y1 + (size_t)slab * NP * CCH;
    int rowbase = mtile * 16 + ((lane >> 4) << 3);
    #pragma unroll
    for (int r = 0; r < 8; ++r)                  // pad rows of L are zero -> Y1 pad rows = 0
        Ys[(rowbase + r) * CCH + col] = acc[r];  // coalesced 64B/instr
}

// ---------------- GEMM2 fused: Z = L*Y1 ; out = Xr*Wc + Y1*W1 + Z*(2W2) + b ----------------
__global__ __launch_bounds__(128)
void k_gemm2_out(const float* __restrict__ Lp, const float* __restrict__ Xr,
                 const float* __restrict__ Y1, const float* __restrict__ Wm,
                 const float* __restrict__ bias, float* __restrict__ out) {
    __shared__ float Lpan[16 * LPS];
    __shared__ float Zt[16 * ZTS];               // Z tile: 16 rows x 64 ch, padded stride
    int mtile = blockIdx.x;
    int slab  = blockIdx.y;
    int bidx  = slab / TT;
    int tt    = slab % TT;
    int tid  = threadIdx.x;
    int wave = tid >> 5;
    int lane = tid & 31;
    int m  = lane & 15;
    int kh = (lane >> 4) * 2;

    for (int i = tid; i < 16 * NP; i += 128)
        Lpan[(i / NP) * LPS + (i % NP)] = Lp[(mtile * 16 + i / NP) * NP + (i % NP)];
    __syncthreads();

    const float* Ys = Y1 + (size_t)slab * NP * CCH;
    const float* Xs = Xr + (size_t)slab * NP * CCH;

    // ---- phase 1: Z c-tile (wave -> ctile) ----
    {
        int col = wave * 16 + m;
        v8f z = {0.f,0.f,0.f,0.f,0.f,0.f,0.f,0.f};
        #pragma unroll 4
        for (int k = 0; k < NP; k += 4) {
            v2f a  = lds_pair(&Lpan[m * LPS + k + kh]);
            v2f bb;
            bb.x = Ys[(k + kh) * CCH + col];
            bb.y = Ys[(k + kh + 1) * CCH + col];
            z    = wmma4(a, bb, z);
        }
        int rb = (lane >> 4) << 3;
        #pragma unroll
        for (int r = 0; r < 8; ++r)
            Zt[(rb + r) * ZTS + col] = z[r];     // conflict-free with ZTS=68
    }
    __syncthreads();

    // ---- phase 2: output d-tile (wave -> dtile), K = 64 channels ----
    int d  = wave * 16 + m;                      // output channel
    float bv = bias[d];
    v8f acc = {bv,bv,bv,bv,bv,bv,bv,bv};         // bias broadcast in C operand
    int mrow = mtile * 16 + m;                   // global A row (padded rows are zero)

    const float* Wc  = Wm;                       // W0 - W2
    const float* W1m = Wm + CCH * CCH;           // W1
    const float* W2s = Wm + 2 * CCH * CCH;       // 2*W2

    #pragma unroll 4
    for (int k = 0; k < CCH; k += 4) {           // X * (W0 - W2)
        v2f a, bb;
        a.x  = Xs[mrow * CCH + k + kh];          // contiguous pair -> b64
        a.y  = Xs[mrow * CCH + k + kh + 1];
        bb.x = Wc[(k + kh) * CCH + d];
        bb.y = Wc[(k + kh + 1) * CCH + d];
        acc  = wmma4(a, bb, acc);
    }
    #pragma unroll 4
    for (int k = 0; k < CCH; k += 4) {           // Y1 * W1
        v2f a, bb;
        a.x  = Ys[mrow * CCH + k + kh];
        a.y  = Ys[mrow * CCH + k + kh + 1];
        bb.x = W1m[(k + kh) * CCH + d];
        bb.y = W1m[(k + kh + 1) * CCH + d];
        acc  = wmma4(a, bb, acc);
    }
    #pragma unroll 4
    for (int k = 0; k < CCH; k += 4) {           // Z * (2*W2)
        v2f a  = lds_pair(&Zt[m * ZTS + k + kh]);
        v2f bb;
        bb.x = W2s[(k + kh) * CCH + d];
        bb.y = W2s[(k + kh + 1) * CCH + d];
        acc  = wmma4(a, bb, acc);
    }

    // ---- store back to out[B, C, N, T] ----
    int rb = mtile * 16 + ((lane >> 4) << 3);
    #pragma unroll
    for (int r = 0; r < 8; ++r) {
        int n = rb + r;
        if (n < NNODES)
            out[((size_t)(bidx * CCH + d) * NNODES + n) * TT + tt] = acc[r];
    }
}

// ---------------- launch ----------------
extern "C" void kernel_launch(void* const* d_in, const int* in_sizes, int n_in,
                              void* d_out, int out_size, void* d_ws, size_t ws_size,
                              hipStream_t stream) {
    const float* x    = (const float*)d_in[0];   // [64,64,325,12]
    const float* adj  = (const float*)d_in[1];   // [325,325]
    const float* W    = (const float*)d_in[2];   // [3,64,64]
    const float* bias = (const float*)d_in[3];   // [64]
    float* out = (float*)d_out;

    char* ws = (char*)d_ws;
    float* Lp   = (float*)(ws);                  // 336*336*4      = 451,584 B
    float* dinv = (float*)(ws + 460800);         // 325 floats
    float* Wm   = (float*)(ws + 524288);         // 3*64*64*4      = 49,152 B
    float* Xr   = (float*)(ws + 1048576);        // 768*336*64*4   = 66,060,288 B
    float* Y1   = (float*)(ws + 67108864);       // 768*336*64*4   = 66,060,288 B
    // total workspace used: ~133.2 MB (Xr + Y1 + L stay L2-resident: 192MB global L2)

    k_deg   <<<dim3(2),        dim3(192), 0, stream>>>(adj, dinv);
    k_lap   <<<dim3(NP),       dim3(352), 0, stream>>>(adj, dinv, Lp);
    k_wprep <<<dim3(16),       dim3(256), 0, stream>>>(W, Wm);
    k_xpose <<<dim3(BB, NP),   dim3(256), 0, stream>>>(x, Xr);
    k_gemm_LX  <<<dim3(MT, BT), dim3(128), 0, stream>>>(Lp, Xr, Y1);
    k_gemm2_out<<<dim3(MT, BT), dim3(128), 0, stream>>>(Lp, Xr, Y1, Wm, bias, out);
}